// YOLOV2Loss_7232724926898
// MI455X (gfx1250) — compile-verified
//
#include <hip/hip_runtime.h>
#include <stdint.h>

// ---- problem constants (from reference setup_inputs) ----
#define GS   19        // grid S
#define SS   361       // S*S
#define BN   5         // anchors
#define NCLS 20        // classes
#define CHP  25        // channels per anchor (C+5)
#define TT   50        // max targets per image
// noobj test: iou > 0.6  <=>  inter > 0.6*union  <=>  inter > 0.375*(pArea + tArea)
#define IOU_K 0.375f

typedef __attribute__((ext_vector_type(2))) float v2f;
typedef __attribute__((ext_vector_type(8))) float v8f;

__device__ __forceinline__ float sigmoidf_(float x) {
  return 1.0f / (1.0f + __expf(-x));
}

__device__ __forceinline__ float iou_box(float cx1, float cy1, float w1, float h1,
                                         float cx2, float cy2, float w2, float h2) {
  float x1a = cx1 - w1 * 0.5f, x1b = cx1 + w1 * 0.5f;
  float y1a = cy1 - h1 * 0.5f, y1b = cy1 + h1 * 0.5f;
  float x2a = cx2 - w2 * 0.5f, x2b = cx2 + w2 * 0.5f;
  float y2a = cy2 - h2 * 0.5f, y2b = cy2 + h2 * 0.5f;
  float iw = fmaxf(fminf(x1b, x2b) - fmaxf(x1a, x2a), 0.0f);
  float ih = fmaxf(fminf(y1b, y2b) - fmaxf(y1a, y2a), 0.0f);
  float inter = iw * ih;
  float uni = w1 * h1 + w2 * h2 - inter;
  return inter / uni;
}

// ---- kernel 1: init owner table (sentinel = TT) and zero the scalar output ----
__global__ void k_init(int* __restrict__ owner, float* __restrict__ out, int ncell) {
  int i = blockIdx.x * blockDim.x + threadIdx.x;
  if (i < ncell) owner[i] = TT;
  if (i == 0) out[0] = 0.0f;
}

// ---- kernel 2: per-target best-anchor + scatter-min owner (matches reference .at[].min) ----
__global__ void k_owner(const float* __restrict__ targets,
                        const float* __restrict__ anchors,
                        int* __restrict__ owner, int bs) {
  int idx = blockIdx.x * blockDim.x + threadIdx.x;   // b*TT + t
  if (idx >= bs * TT) return;
  int b = idx / TT, t = idx % TT;
  const float* tg = targets + (size_t)idx * 5;
  float x = tg[1], y = tg[2], w = tg[3], h = tg[4];
  if (!(x > 0.0f)) return;                           // valid = x > 0
  const float invS = 1.0f / (float)GS;
  int best = 0; float bestr = -1.0f;
#pragma unroll
  for (int a = 0; a < BN; a++) {
    float aw = anchors[2 * a] * invS, ah = anchors[2 * a + 1] * invS;
    float inter = fminf(aw, w) * fminf(ah, h);
    float uni = aw * ah + w * h - inter;
    float r = inter / uni;
    if (r > bestr) { bestr = r; best = a; }          // first-max wins, like argmax
  }
  int ii = min(max((int)(x * (float)GS), 0), GS - 1);
  int jj = min(max((int)(y * (float)GS), 0), GS - 1);
  int cell = ((b * GS + jj) * GS + ii) * BN + best;
  atomicMin(&owner[cell], t);
}

// ---- kernel 3: streaming fused loss. One block per (batch, anchor): 361 cells x 25 ch ----
__launch_bounds__(384, 1)
__global__ void k_main(const float* __restrict__ preds,
                       const float* __restrict__ targets,
                       const float* __restrict__ anchors,
                       const int* __restrict__ owner,
                       float* __restrict__ out, int bs) {
  __shared__ float sh_raw[TT * 5];   // raw targets (cls,x,y,w,h) for obj-cell path
  __shared__ float sh_box[TT * 5];   // precomputed {xa, xb, ya, yb, 0.375*area}
  __shared__ float red[384];         // block-reduction partials

  const int tid = threadIdx.x;
  const int b = blockIdx.x / BN;
  const int a = blockIdx.x % BN;

  // --- CDNA5 async DMA: stage this batch's 250 target floats global->LDS ---
  if (tid < TT * 5) {
    uint32_t lds_addr = (uint32_t)(uintptr_t)(const void*)&sh_raw[tid]; // low 32b = LDS offset
    uint32_t goff = (uint32_t)((b * TT * 5 + tid) * 4);
    asm volatile("global_load_async_to_lds_b32 %0, %1, %2"
                 :: "v"(lds_addr), "v"(goff), "s"(targets) : "memory");
  }
  const float aw_s = anchors[2 * a];
  const float ah_s = anchors[2 * a + 1];
  asm volatile("s_wait_asynccnt 0" ::: "memory");
  __syncthreads();

  // --- precompute target box corners + scaled area (sentinels for invalid) ---
  if (tid < TT) {
    float gx = sh_raw[tid * 5 + 1], gy = sh_raw[tid * 5 + 2];
    float gw = sh_raw[tid * 5 + 3], gh = sh_raw[tid * 5 + 4];
    bool valid = gx > 0.0f;
    sh_box[tid * 5 + 0] = valid ? gx - gw * 0.5f : 1e30f;
    sh_box[tid * 5 + 1] = valid ? gx + gw * 0.5f : -1e30f;
    sh_box[tid * 5 + 2] = valid ? gy - gh * 0.5f : 1e30f;
    sh_box[tid * 5 + 3] = valid ? gy + gh * 0.5f : -1e30f;
    sh_box[tid * 5 + 4] = valid ? IOU_K * gw * gh : 1e30f;   // unreachable threshold if invalid
  }
  __syncthreads();

  float lloss = 0.0f;
  if (tid < SS) {
    const int pos = tid;
    const int i = pos % GS, j = pos / GS;
    const float invS = 1.0f / (float)GS;
    const float* base = preds + ((size_t)(b * (BN * CHP) + a * CHP)) * SS + pos;

    float o[CHP];
#pragma unroll
    for (int c = 0; c < CHP; c++) o[c] = base[(size_t)c * SS];  // coalesced per channel plane

    float tx = sigmoidf_(o[0]);
    float ty = sigmoidf_(o[1]);
    float tw = o[2], th = o[3];
    float conf = sigmoidf_(o[4]);
    float cx = (tx + (float)i) * invS;
    float cy = (ty + (float)j) * invS;
    float pw = __expf(tw) * aw_s * invS;
    float ph = __expf(th) * ah_s * invS;

    const int cell = ((b * GS + j) * GS + i) * BN + a;
    const int town = owner[cell];

    if (town < TT) {
      // ---- object cell: obj/coord/class losses ----
      const float* tg = &sh_raw[town * 5];
      const int tcls = (int)tg[0];
      const float gx = tg[1], gy = tg[2], gw = tg[3], gh = tg[4];

      float iou_real = iou_box(cx, cy, pw, ph, gx, gy, gw, gh);
      float dob = conf - iou_real;
      lloss += 5.0f * dob * dob;                       // OBJ_SCALE

      float ex = gx * (float)GS - (float)i;
      float ey = gy * (float)GS - (float)j;
      float ew = __logf(gw * (float)GS / aw_s);
      float eh = __logf(gh * (float)GS / ah_s);
      float sc = 2.0f - gw * gh;
      float d0 = (tx - ex) * sc, d1 = (ty - ey) * sc;
      float d2 = (tw - ew) * sc, d3 = (th - eh) * sc;
      lloss += d0 * d0 + d1 * d1 + d2 * d2 + d3 * d3;  // COORD_SCALE = 1

      // softmax only where it's consumed (obj cells, ~3%)
      float m = o[5];
#pragma unroll
      for (int c = 6; c < CHP; c++) m = fmaxf(m, o[c]);
      float e[NCLS]; float sum = 0.0f;
#pragma unroll
      for (int c = 0; c < NCLS; c++) { e[c] = __expf(o[5 + c] - m); sum += e[c]; }
      float inv = 1.0f / sum;
      float cl = 0.0f;
#pragma unroll
      for (int c = 0; c < NCLS; c++) {
        float p = e[c] * inv;
        float d = p - ((c == tcls) ? 1.0f : 0.0f);
        cl += d * d;
      }
      lloss += cl;                                     // CLASS_SCALE = 1
    } else {
      // ---- background cell: noobj + prior losses ----
      // division-free, branchless: any(inter > 0.375*(pArea + tArea)) ?
      float xa_p = cx - pw * 0.5f, xb_p = cx + pw * 0.5f;
      float ya_p = cy - ph * 0.5f, yb_p = cy + ph * 0.5f;
      float c_p = IOU_K * pw * ph;
      int hit = 0;
#pragma unroll 5
      for (int t = 0; t < TT; t++) {                   // wave-uniform LDS broadcast reads
        float iw = fmaxf(fminf(xb_p, sh_box[t * 5 + 1]) - fmaxf(xa_p, sh_box[t * 5 + 0]), 0.0f);
        float ih = fmaxf(fminf(yb_p, sh_box[t * 5 + 3]) - fmaxf(ya_p, sh_box[t * 5 + 2]), 0.0f);
        hit |= (iw * ih > c_p + sh_box[t * 5 + 4]);
      }
      if (!hit) lloss += conf * conf;                  // NOOBJ_SCALE = 1
      float p0 = tx - 0.5f, p1 = ty - 0.5f;
      lloss += 0.01f * (p0 * p0 + p1 * p1 + tw * tw + th * th); // PRIOR_SCALE
    }
  }

  red[tid] = lloss;
  __syncthreads();

  // ---- block reduction: wave 0 sums 384 partials with f32 WMMA (A = ones 16x4) ----
  // D = ones(16x4) x B(4x16) + C  =>  every row of D holds the 16 column-sums of B.
  // Layout-agnostic for a total-sum: sum(acc[0] over 32 lanes) == 2 * sum(all B elems).
  if (tid < 32) {
    v8f acc = {};
    v2f ones; ones.x = 1.0f; ones.y = 1.0f;
#pragma unroll
    for (int it = 0; it < 6; it++) {                   // 6 x 64 = 384 partials
      v2f bb;
      bb.x = red[it * 64 + 2 * tid];
      bb.y = red[it * 64 + 2 * tid + 1];
      acc = __builtin_amdgcn_wmma_f32_16x16x4_f32(
          /*neg_a=*/false, ones, /*neg_b=*/false, bb,
          /*c_mod=*/(short)0, acc, /*reuse_a=*/false, /*reuse_b=*/false);
    }
    float v = acc[0];
#pragma unroll
    for (int off = 16; off > 0; off >>= 1) v += __shfl_xor(v, off);
    if (tid == 0) atomicAdd(out, v * (0.5f / (float)bs)); // halve row duplication, /bs
  }
}

extern "C" void kernel_launch(void* const* d_in, const int* in_sizes, int n_in,
                              void* d_out, int out_size, void* d_ws, size_t ws_size,
                              hipStream_t stream) {
  const float* preds   = (const float*)d_in[0];
  const float* targets = (const float*)d_in[1];
  const float* anchors = (const float*)d_in[2];
  float* out = (float*)d_out;

  const int bs = in_sizes[0] / (BN * CHP * SS);
  const int ncell = bs * SS * BN;
  int* owner = (int*)d_ws;   // ncell ints of scratch

  k_init<<<(ncell + 255) / 256, 256, 0, stream>>>(owner, out, ncell);
  k_owner<<<(bs * TT + 255) / 256, 256, 0, stream>>>(targets, anchors, owner, bs);
  k_main<<<bs * BN, 384, 0, stream>>>(preds, targets, anchors, owner, out, bs);
}